// GAT_4904852652367
// MI455X (gfx1250) — compile-verified
//
#include <hip/hip_runtime.h>

#define N_NODES 50000
#define N_EDGES 800000
#define EN (N_EDGES + N_NODES)
#define NFEAT 9
#define VOCAB 128
#define DIM 96
#define NGRAPH 512
#define NOUT 10
#define NLAYER 4
#define NEG_SLOPE 0.2f

// K-pair interleaved W layout in LDS: pitch per k-pair row (in floats).
// 192 payload floats (96 cols x 2) padded to 224 (224 mod 64 == 32 ->
// the two half-waves, which read adjacent k-pairs, hit disjoint bank halves).
#define WPITCH 224

typedef float v2f __attribute__((ext_vector_type(2)));
typedef float v8f __attribute__((ext_vector_type(8)));

// ---------------- helpers ----------------

__device__ __forceinline__ unsigned fenc(float f) {
    unsigned u = __float_as_uint(f);
    return (u >> 31) ? ~u : (u | 0x80000000u);
}
__device__ __forceinline__ float fdec(unsigned u) {
    return __uint_as_float((u >> 31) ? (u & 0x7FFFFFFFu) : ~u);
}
__device__ __forceinline__ void edge_sd(const int* __restrict__ ei, int i, int& s, int& d) {
    if (i < N_EDGES) { s = ei[i]; d = ei[N_EDGES + i]; }
    else             { s = i - N_EDGES; d = s; }
}

// ---------------- kernels ----------------

__global__ void k_zero(float* __restrict__ p, int n) {
    for (int i = blockIdx.x * blockDim.x + threadIdx.x; i < n; i += gridDim.x * blockDim.x)
        p[i] = 0.0f;
}

// h[n,d] = sum_f emb[f, x[n,f], d]
__global__ void k_atom_encode(const int* __restrict__ x, const float* __restrict__ emb,
                              float* __restrict__ h) {
    int idx = blockIdx.x * blockDim.x + threadIdx.x;
    if (idx >= N_NODES * DIM) return;
    int n = idx / DIM, d = idx % DIM;
    float s = 0.0f;
#pragma unroll
    for (int f = 0; f < NFEAT; ++f) {
        int tok = x[n * NFEAT + f];
        s += emb[((size_t)f * VOCAB + tok) * DIM + d];
    }
    h[idx] = s;
}

// hw = h @ W via V_WMMA_F32_16X16X4_F32. One wave -> 16x96 output tile.
// B fragments come from a k-pair-interleaved LDS image of W so each fragment
// is a single aligned ds_load_b64 landing in an even VGPR pair (no shuffles).
__global__ __launch_bounds__(256)
void k_gemm_wmma(const float* __restrict__ h, const float* __restrict__ W,
                 float* __restrict__ hw) {
    __shared__ float sW2[(DIM / 2) * WPITCH];   // 48 * 224 * 4B = 43008 B
    for (int i = threadIdx.x; i < DIM * DIM; i += 256) {
        int k = i / DIM, c = i % DIM;
        sW2[(k >> 1) * WPITCH + (c << 1) + (k & 1)] = W[i];
    }
    __syncthreads();

    const int wave = threadIdx.x >> 5;
    const int lane = threadIdx.x & 31;
    const int tile = blockIdx.x * 8 + wave;
    if (tile >= N_NODES / 16) return;          // 50000/16 = 3125 exact tiles
    const int rowBase = tile * 16;
    const int m = lane & 15;                   // A row / B,C col
    const int kp_off = lane >> 4;              // lanes 16..31 hold K+2,K+3

    v8f acc[6] = {};
    const float* arow = h + (size_t)(rowBase + m) * DIM;

    for (int k0 = 0; k0 < DIM; k0 += 4) {
        const int kk = k0 + (kp_off << 1);
        float2 av = *(const float2*)(arow + kk);       // 8B aligned (kk even)
        v2f a; a.x = av.x; a.y = av.y;
        const float* brow = &sW2[(size_t)(kk >> 1) * WPITCH];
#pragma unroll
        for (int t = 0; t < 6; ++t) {
            v2f b = *(const v2f*)(brow + ((t * 16 + m) << 1));  // {W[kk][c], W[kk+1][c]}
            acc[t] = __builtin_amdgcn_wmma_f32_16x16x4_f32(
                false, a, false, b, (short)0, acc[t], false, false);
        }
    }

    const int rowOff = (lane >> 4) << 3;       // C/D: lanes 16..31 hold rows 8..15
#pragma unroll
    for (int t = 0; t < 6; ++t)
#pragma unroll
        for (int v = 0; v < 8; ++v)
            hw[(size_t)(rowBase + rowOff + v) * DIM + t * 16 + m] = acc[t][v];
}

// sc_s[n] = hw[n,:].a_s ; sc_d[n] = hw[n,:].a_d  (one wave per node)
__global__ void k_scores(const float* __restrict__ hw, const float* __restrict__ as,
                         const float* __restrict__ ad, float* __restrict__ scs,
                         float* __restrict__ scd) {
    int node = (blockIdx.x * blockDim.x + threadIdx.x) >> 5;
    int lane = threadIdx.x & 31;
    if (node >= N_NODES) return;
    float ss = 0.0f, sd = 0.0f;
#pragma unroll
    for (int j = lane; j < DIM; j += 32) {
        float v = hw[(size_t)node * DIM + j];
        ss += v * as[j];
        sd += v * ad[j];
    }
#pragma unroll
    for (int off = 16; off; off >>= 1) {
        ss += __shfl_xor(ss, off, 32);
        sd += __shfl_xor(sd, off, 32);
    }
    if (lane == 0) { scs[node] = ss; scd[node] = sd; }
}

// pass 1: e = leaky_relu(sc_s[src]+sc_d[dst]); segment max over dst
__global__ void k_edge_max(const int* __restrict__ ei, const float* __restrict__ scs,
                           const float* __restrict__ scd, float* __restrict__ ebuf,
                           unsigned* __restrict__ emax) {
    int i = blockIdx.x * blockDim.x + threadIdx.x;
    if (i >= EN) return;
    int s, d; edge_sd(ei, i, s, d);
    float e = scs[s] + scd[d];
    e = (e > 0.0f) ? e : NEG_SLOPE * e;
    ebuf[i] = e;
    atomicMax(&emax[d], fenc(e));
}

// pass 2: ee = exp(e - emax[dst]); segment sum over dst (ebuf overwritten in place)
__global__ void k_edge_exp(const int* __restrict__ ei, float* __restrict__ ebuf,
                           const unsigned* __restrict__ emax, float* __restrict__ denom) {
    int i = blockIdx.x * blockDim.x + threadIdx.x;
    if (i >= EN) return;
    int s, d; edge_sd(ei, i, s, d);
    float ee = __expf(ebuf[i] - fdec(emax[d]));
    ebuf[i] = ee;
    atomicAdd(&denom[d], ee);
}

// pass 3: acc[dst] += (ee/denom[dst]) * hw[src]  (one wave per edge)
__global__ void k_edge_aggr(const int* __restrict__ ei, const float* __restrict__ ebuf,
                            const float* __restrict__ denom, const float* __restrict__ hw,
                            float* __restrict__ acc) {
    int e = (blockIdx.x * blockDim.x + threadIdx.x) >> 5;
    int lane = threadIdx.x & 31;
    if (e >= EN) return;
    int s, d; edge_sd(ei, e, s, d);
    const float* srow = hw + (size_t)s * DIM;
    __builtin_prefetch(srow + lane, 0, 3);             // global_prefetch_b8: start row fetch
    float coef = ebuf[e] / denom[d];
#pragma unroll
    for (int j = lane; j < DIM; j += 32)
        atomicAdd(&acc[(size_t)d * DIM + j], coef * srow[j]);
}

// in-place: h = relu(h + b)
__global__ void k_bias_relu(float* __restrict__ h, const float* __restrict__ b) {
    int idx = blockIdx.x * blockDim.x + threadIdx.x;
    if (idx >= N_NODES * DIM) return;
    float v = h[idx] + b[idx % DIM];
    h[idx] = v > 0.0f ? v : 0.0f;
}

// mean-pool inputs: pooled[g] += h[n]; cnt[g] += 1 (one wave per node)
__global__ void k_pool(const float* __restrict__ h, const int* __restrict__ batch,
                       float* __restrict__ pooled, float* __restrict__ cnt) {
    int node = (blockIdx.x * blockDim.x + threadIdx.x) >> 5;
    int lane = threadIdx.x & 31;
    if (node >= N_NODES) return;
    int g = batch[node];
#pragma unroll
    for (int j = lane; j < DIM; j += 32)
        atomicAdd(&pooled[(size_t)g * DIM + j], h[(size_t)node * DIM + j]);
    if (lane == 0) atomicAdd(&cnt[g], 1.0f);
}

// out[g,o] = (pooled[g]/max(cnt,1)) . lin_w[:,o] + lin_b[o]
__global__ void k_head(const float* __restrict__ pooled, const float* __restrict__ cnt,
                       const float* __restrict__ lw, const float* __restrict__ lb,
                       float* __restrict__ out) {
    int i = blockIdx.x * blockDim.x + threadIdx.x;
    if (i >= NGRAPH * NOUT) return;
    int g = i / NOUT, o = i % NOUT;
    float c = fmaxf(cnt[g], 1.0f);
    float s = lb[o];
    for (int d = 0; d < DIM; ++d)
        s += (pooled[(size_t)g * DIM + d] / c) * lw[d * NOUT + o];
    out[i] = s;
}

// ---------------- launch ----------------

extern "C" void kernel_launch(void* const* d_in, const int* in_sizes, int n_in,
                              void* d_out, int out_size, void* d_ws, size_t ws_size,
                              hipStream_t stream) {
    const int*   x       = (const int*)d_in[0];
    const int*   ei      = (const int*)d_in[1];
    const int*   batch   = (const int*)d_in[2];
    const float* emb     = (const float*)d_in[3];
    const float* Ws      = (const float*)d_in[4];
    const float* att_src = (const float*)d_in[5];
    const float* att_dst = (const float*)d_in[6];
    const float* biases  = (const float*)d_in[7];
    const float* lin_w   = (const float*)d_in[8];
    const float* lin_b   = (const float*)d_in[9];
    float* out = (float*)d_out;

    // workspace layout (floats); h doubles as the aggregation accumulator
    float*    h      = (float*)d_ws;                  // N*D
    float*    hw     = h   + (size_t)N_NODES * DIM;   // N*D
    float*    scs    = hw  + (size_t)N_NODES * DIM;   // N
    float*    scd    = scs + N_NODES;                 // N
    unsigned* emax   = (unsigned*)(scd + N_NODES);    // N
    float*    denom  = (float*)(emax + N_NODES);      // N
    float*    ebuf   = denom + N_NODES;               // EN
    float*    pooled = ebuf + EN;                     // G*D
    float*    cnt    = pooled + (size_t)NGRAPH * DIM; // G

    const int nd = N_NODES * DIM;

    k_atom_encode<<<(nd + 255) / 256, 256, 0, stream>>>(x, emb, h);

    for (int l = 0; l < NLAYER; ++l) {
        const float* W  = Ws + (size_t)l * DIM * DIM;
        const float* as = att_src + l * DIM;
        const float* ad = att_dst + l * DIM;
        const float* b  = biases  + l * DIM;

        k_gemm_wmma<<<(N_NODES / 16 + 7) / 8, 256, 0, stream>>>(h, W, hw);
        k_zero<<<(nd + 255) / 256, 256, 0, stream>>>(h, nd);                 // acc = 0
        k_zero<<<(2 * N_NODES + 255) / 256, 256, 0, stream>>>((float*)emax, 2 * N_NODES);
        k_scores<<<(N_NODES * 32 + 255) / 256, 256, 0, stream>>>(hw, as, ad, scs, scd);
        k_edge_max<<<(EN + 255) / 256, 256, 0, stream>>>(ei, scs, scd, ebuf, emax);
        k_edge_exp<<<(EN + 255) / 256, 256, 0, stream>>>(ei, ebuf, emax, denom);
        k_edge_aggr<<<((size_t)EN * 32 + 255) / 256, 256, 0, stream>>>(ei, ebuf, denom, hw, h);
        k_bias_relu<<<(nd + 255) / 256, 256, 0, stream>>>(h, b);
    }

    k_zero<<<(NGRAPH * DIM + NGRAPH + 255) / 256, 256, 0, stream>>>(pooled, NGRAPH * DIM + NGRAPH);
    k_pool<<<(N_NODES * 32 + 255) / 256, 256, 0, stream>>>(h, batch, pooled, cnt);
    k_head<<<(NGRAPH * NOUT + 255) / 256, 256, 0, stream>>>(pooled, cnt, lin_w, lin_b, out);
}